// NystromAttention2D_78134045048865
// MI455X (gfx1250) — compile-verified
//
#include <hip/hip_runtime.h>
#include <cstddef>
#include <cstdint>

#define HEADS  8
#define DHEAD  64
#define NCLUST 512
#define NTOK   16384
#define DIM    512
#define QKVLD  1536
#define KW     33

typedef __attribute__((ext_vector_type(2))) float v2f;
typedef __attribute__((ext_vector_type(8))) float v8f;

__device__ __forceinline__ v8f wmma4(v2f a, v2f b, v8f c) {
  // D(16x16,f32) = A(16x4,f32) * B(4x16,f32) + C
  return __builtin_amdgcn_wmma_f32_16x16x4_f32(false, a, false, b, (short)0, c, false, false);
}

__device__ __forceinline__ void atomicMaxF(float* addr, float v) {
  atomicMax((unsigned int*)addr, __float_as_uint(v)); // valid: values >= 0
}

// ---- CDNA5 async global->LDS DMA (ASYNCcnt path, ISA cdna5 ch.10/15.18) ----
__device__ __forceinline__ unsigned lds32(const void* p) {
  // flat LDS aperture: addr[31:0] is the wave-relative LDS byte offset
  return (unsigned)(unsigned long long)p;
}
__device__ __forceinline__ void async_b128(const float* g, unsigned l) {
  asm volatile("global_load_async_to_lds_b128 %0, %1, off"
               :: "v"(l), "v"(g) : "memory");
}
__device__ __forceinline__ void async_b32(const float* g, unsigned l) {
  asm volatile("global_load_async_to_lds_b32 %0, %1, off"
               :: "v"(l), "v"(g) : "memory");
}
__device__ __forceinline__ void wait_async0() {
  asm volatile("s_wait_asynccnt 0x0" ::: "memory");
}

// ---------------------------------------------------------------- sort
__global__ void hist_kernel(const int* __restrict__ labels, int* __restrict__ counts) {
  int i = blockIdx.x * 256 + threadIdx.x;
  if (i < NTOK) atomicAdd(&counts[labels[i]], 1);
}

__global__ void scan_kernel(const int* __restrict__ counts, int* __restrict__ offs) {
  __shared__ int tmp[NCLUST];
  int t = threadIdx.x;
  tmp[t] = counts[t];
  __syncthreads();
  for (int o = 1; o < NCLUST; o <<= 1) {
    int v = (t >= o) ? tmp[t - o] : 0;
    __syncthreads();
    tmp[t] += v;
    __syncthreads();
  }
  offs[t] = tmp[t] - counts[t]; // exclusive
}

__global__ void stable_sort_kernel(const int* __restrict__ labels,
                                   const int* __restrict__ offs,
                                   int* __restrict__ idx) {
  int c = blockIdx.x * 256 + threadIdx.x;
  if (c >= NCLUST) return;
  int p = offs[c];
  for (int i = 0; i < NTOK; ++i)
    if (labels[i] == c) idx[p++] = i;
}

// ---------------------------------------------------------------- big GEMM
// C[r][j] = sum_k A[gather(r)][k] * W[j][k]  (+bias), rows optionally scattered.
// Double-buffered async global->LDS tile staging.
__global__ __launch_bounds__(256)
void gemm_big(const float* __restrict__ A, int lda,
              const int* __restrict__ gidx,
              const float* __restrict__ W,   // N x K row-major (weights (out,in))
              float* __restrict__ C, int ldc,
              const int* __restrict__ sidx,
              const float* __restrict__ bias,
              int K) {
  __shared__ float sA[2][128][20]; // row stride 80B (16B aligned), conflict-free mod 64
  __shared__ float sB[2][16][68];  // row stride 272B (16B aligned)
  int tid = threadIdx.x;
  int lane = tid & 31, wv = tid >> 5;
  int l15 = lane & 15, hi = lane >> 4;
  int n0 = blockIdx.x * 64;
  int m0 = blockIdx.y * 128;

  // A tile: 128x16 floats = 512 b128 groups; this thread owns 2 groups.
  int rA0 = tid >> 2;
  int rA1 = rA0 + 64;
  int kqA = (tid & 3) * 4;
  int growA0 = gidx ? gidx[m0 + rA0] : (m0 + rA0);
  int growA1 = gidx ? gidx[m0 + rA1] : (m0 + rA1);

  auto issueA = [&](int k0, int buf) {
    async_b128(A + (size_t)growA0 * lda + k0 + kqA, lds32(&sA[buf][rA0][kqA]));
    async_b128(A + (size_t)growA1 * lda + k0 + kqA, lds32(&sA[buf][rA1][kqA]));
  };
  // B tile: transpose load from W (strided in global) -> 4 b32 per thread.
  auto issueB = [&](int k0, int buf) {
#pragma unroll
    for (int q = 0; q < 4; ++q) {
      int e = tid + 256 * q;
      int j = e & 63, kk = e >> 6;
      async_b32(W + (size_t)(n0 + j) * K + k0 + kk, lds32(&sB[buf][kk][j]));
    }
  };

  issueA(0, 0);
  issueB(0, 0);

  v8f acc[4] = {};
  int ar = wv * 16 + l15;
  for (int k0 = 0; k0 < K; k0 += 16) {
    int cur = (k0 >> 4) & 1;
    wait_async0();
    __syncthreads();
    if (k0 + 16 < K) { issueA(k0 + 16, cur ^ 1); issueB(k0 + 16, cur ^ 1); }
#pragma unroll
    for (int kk = 0; kk < 16; kk += 4) {
      int kb = kk + hi * 2;
      v2f a; a.x = sA[cur][ar][kb]; a.y = sA[cur][ar][kb + 1];
#pragma unroll
      for (int t = 0; t < 4; ++t) {
        v2f b; b.x = sB[cur][kb][t * 16 + l15]; b.y = sB[cur][kb + 1][t * 16 + l15];
        acc[t] = wmma4(a, b, acc[t]);
      }
    }
  }
#pragma unroll
  for (int t = 0; t < 4; ++t) {
    int col = n0 + t * 16 + l15;
    float bv = bias ? bias[col] : 0.f;
#pragma unroll
    for (int i = 0; i < 8; ++i) {
      int r = m0 + wv * 16 + i + hi * 8;
      int gr = sidx ? sidx[r] : r;
      C[(size_t)gr * ldc + col] = acc[t][i] + bv;
    }
  }
}

// ---------------------------------------------------------------- batched 512^2 GEMM (pinv)
// D = alpha*(A@B) + diag*I, per head (blockIdx.z). A:MxK row-major, B:KxN row-major, K=512.
__global__ __launch_bounds__(256)
void gemm512(const float* __restrict__ A, int lda, long sA_,
             const float* __restrict__ B, int ldb, long sB_,
             float* __restrict__ D, int ldd, long sD_,
             float alpha, float diag) {
  __shared__ float sA[2][128][20];
  __shared__ float sB[2][16][68];
  int h = blockIdx.z;
  const float* Ah = A + (size_t)h * sA_;
  const float* Bh = B + (size_t)h * sB_;
  float* Dh = D + (size_t)h * sD_;
  int tid = threadIdx.x;
  int lane = tid & 31, wv = tid >> 5;
  int l15 = lane & 15, hi = lane >> 4;
  int n0 = blockIdx.x * 64;
  int m0 = blockIdx.y * 128;

  int rA0 = tid >> 2;
  int rA1 = rA0 + 64;
  int kqA = (tid & 3) * 4;
  // B tile: 16x64 floats = 256 b128 groups; one per thread (coalesced, B row-major).
  int kB = tid >> 4;
  int jqB = (tid & 15) * 4;

  auto issueA = [&](int k0, int buf) {
    async_b128(Ah + (size_t)(m0 + rA0) * lda + k0 + kqA, lds32(&sA[buf][rA0][kqA]));
    async_b128(Ah + (size_t)(m0 + rA1) * lda + k0 + kqA, lds32(&sA[buf][rA1][kqA]));
  };
  auto issueB = [&](int k0, int buf) {
    async_b128(Bh + (size_t)(k0 + kB) * ldb + n0 + jqB, lds32(&sB[buf][kB][jqB]));
  };

  issueA(0, 0);
  issueB(0, 0);

  v8f acc[4] = {};
  int ar = wv * 16 + l15;
  for (int k0 = 0; k0 < 512; k0 += 16) {
    int cur = (k0 >> 4) & 1;
    wait_async0();
    __syncthreads();
    if (k0 + 16 < 512) { issueA(k0 + 16, cur ^ 1); issueB(k0 + 16, cur ^ 1); }
#pragma unroll
    for (int kk = 0; kk < 16; kk += 4) {
      int kb = kk + hi * 2;
      v2f a; a.x = sA[cur][ar][kb]; a.y = sA[cur][ar][kb + 1];
#pragma unroll
      for (int t = 0; t < 4; ++t) {
        v2f b; b.x = sB[cur][kb][t * 16 + l15]; b.y = sB[cur][kb + 1][t * 16 + l15];
        acc[t] = wmma4(a, b, acc[t]);
      }
    }
  }
#pragma unroll
  for (int t = 0; t < 4; ++t) {
    int col = n0 + t * 16 + l15;
#pragma unroll
    for (int i = 0; i < 8; ++i) {
      int r = m0 + wv * 16 + i + hi * 8;
      float v = alpha * acc[t][i];
      if (r == col) v += diag;
      Dh[(size_t)r * ldd + col] = v;
    }
  }
}

// ---------------------------------------------------------------- landmarks
__global__ void landmarks_kernel(const float* __restrict__ qkv,
                                 const int* __restrict__ offs,
                                 const int* __restrict__ counts,
                                 float* __restrict__ ql, float* __restrict__ kl) {
  int c = blockIdx.x, h = blockIdx.y, d = threadIdx.x;
  int st = offs[c], cnt = counts[c];
  float sq = 0.f, sk = 0.f;
  for (int r = st; r < st + cnt; ++r) {
    sq += qkv[(size_t)r * QKVLD + h * 64 + d];
    sk += qkv[(size_t)r * QKVLD + 512 + h * 64 + d];
  }
  float inv = 1.0f / (float)(cnt > 1 ? cnt : 1);
  ql[(size_t)h * 32768 + c * 64 + d] = sq * inv;
  kl[(size_t)h * 32768 + c * 64 + d] = sk * inv;
}

// ---------------------------------------------------------------- a2 = softmax(ql @ kl^T)
__global__ __launch_bounds__(256)
void a2_kernel(const float* __restrict__ ql, const float* __restrict__ kl,
               float* __restrict__ a2) {
  __shared__ float sc[16][513];
  __shared__ float red[16][16];
  __shared__ float rm[16], rl[16];
  int h = blockIdx.y, rb = blockIdx.x;
  int tid = threadIdx.x, lane = tid & 31, wv = tid >> 5;
  int l15 = lane & 15, hi = lane >> 4;
  const float* qh = ql + (size_t)h * 32768;
  const float* kh = kl + (size_t)h * 32768;
  int r0 = rb * 16, c0 = wv * 64;
  v8f acc[4] = {};
#pragma unroll
  for (int kk = 0; kk < 64; kk += 4) {
    int kb = kk + hi * 2;
    v2f a; a.x = qh[(r0 + l15) * 64 + kb]; a.y = qh[(r0 + l15) * 64 + kb + 1];
#pragma unroll
    for (int t = 0; t < 4; ++t) {
      int col = c0 + t * 16 + l15;
      v2f b; b.x = kh[col * 64 + kb]; b.y = kh[col * 64 + kb + 1];
      acc[t] = wmma4(a, b, acc[t]);
    }
  }
#pragma unroll
  for (int t = 0; t < 4; ++t)
#pragma unroll
    for (int i = 0; i < 8; ++i)
      sc[i + hi * 8][c0 + t * 16 + l15] = acc[t][i];
  __syncthreads();
  int row = tid >> 4, s = tid & 15;
  float m = -1e30f;
  for (int j = s; j < 512; j += 16) m = fmaxf(m, sc[row][j]);
  red[row][s] = m;
  __syncthreads();
  if (s == 0) { float mm = -1e30f; for (int t = 0; t < 16; ++t) mm = fmaxf(mm, red[row][t]); rm[row] = mm; }
  __syncthreads();
  float mm = rm[row];
  float ps = 0.f;
  for (int j = s; j < 512; j += 16) ps += __expf(sc[row][j] - mm);
  red[row][s] = ps;
  __syncthreads();
  if (s == 0) { float ss = 0.f; for (int t = 0; t < 16; ++t) ss += red[row][t]; rl[row] = 1.0f / ss; }
  __syncthreads();
  float inv = rl[row];
  for (int j = s; j < 512; j += 16)
    a2[(size_t)h * 262144 + (size_t)(r0 + row) * 512 + j] = __expf(sc[row][j] - mm) * inv;
}

// ---------------------------------------------------------------- pinv scaling
__global__ void rowsum_max(const float* __restrict__ a2, float* __restrict__ dst) {
  __shared__ float red[256];
  int i = blockIdx.x, h = blockIdx.y, t = threadIdx.x;
  const float* rp = a2 + (size_t)h * 262144 + (size_t)i * 512;
  red[t] = fabsf(rp[t]) + fabsf(rp[t + 256]);
  __syncthreads();
  for (int o = 128; o > 0; o >>= 1) { if (t < o) red[t] += red[t + o]; __syncthreads(); }
  if (t == 0) atomicMaxF(dst, red[0]);
}

__global__ void colsum_max(const float* __restrict__ a2, float* __restrict__ dst) {
  __shared__ float red[256];
  int j = blockIdx.x, h = blockIdx.y, t = threadIdx.x;
  const float* cp = a2 + (size_t)h * 262144 + j;
  red[t] = fabsf(cp[(size_t)t * 512]) + fabsf(cp[(size_t)(t + 256) * 512]);
  __syncthreads();
  for (int o = 128; o > 0; o >>= 1) { if (t < o) red[t] += red[t + o]; __syncthreads(); }
  if (t == 0) atomicMaxF(dst, red[0]);
}

__global__ void zinit_kernel(const float* __restrict__ a2, const float* __restrict__ scal,
                             float* __restrict__ z) {
  float s = 1.0f / (scal[0] * scal[1]);
  int gid = blockIdx.x * 256 + threadIdx.x;
  int h = blockIdx.y;
  int i = gid >> 9, j = gid & 511;
  z[(size_t)h * 262144 + (size_t)i * 512 + j] = a2[(size_t)h * 262144 + (size_t)j * 512 + i] * s;
}

__global__ void elem7i_kernel(const float* __restrict__ xz, float* __restrict__ t0) {
  int gid = blockIdx.x * 256 + threadIdx.x;
  int h = blockIdx.y;
  int i = gid >> 9, j = gid & 511;
  float d = (i == j) ? 7.0f : 0.0f;
  t0[(size_t)h * 262144 + gid] = d - xz[(size_t)h * 262144 + gid];
}

// ---------------------------------------------------------------- sim3 two-pass (streaming softmax over n)
__global__ __launch_bounds__(256)
void sim3_stats_kernel(const float* __restrict__ ql, const float* __restrict__ qkv,
                       float* __restrict__ m3, float* __restrict__ l3) {
  __shared__ float sc[16][513];
  __shared__ float red[16][16];
  __shared__ float rm[16], rl[16], mnew[16];
  int h = blockIdx.y, rb = blockIdx.x;
  int tid = threadIdx.x, lane = tid & 31, wv = tid >> 5;
  int l15 = lane & 15, hi = lane >> 4;
  const float* qh = ql + (size_t)h * 32768;
  int r0 = rb * 16;
  int row = tid >> 4, s = tid & 15;
  if (tid < 16) { rm[tid] = -1e30f; rl[tid] = 0.f; }
  __syncthreads();
  for (int jc = 0; jc < NTOK; jc += 512) {
    v8f acc[4] = {};
#pragma unroll
    for (int kk = 0; kk < 64; kk += 4) {
      int kb = kk + hi * 2;
      v2f a; a.x = qh[(r0 + l15) * 64 + kb]; a.y = qh[(r0 + l15) * 64 + kb + 1];
#pragma unroll
      for (int t = 0; t < 4; ++t) {
        int col = jc + wv * 64 + t * 16 + l15;
        v2f b;
        b.x = qkv[(size_t)col * QKVLD + 512 + h * 64 + kb];
        b.y = qkv[(size_t)col * QKVLD + 512 + h * 64 + kb + 1];
        acc[t] = wmma4(a, b, acc[t]);
      }
    }
#pragma unroll
    for (int t = 0; t < 4; ++t)
#pragma unroll
      for (int i = 0; i < 8; ++i)
        sc[i + hi * 8][wv * 64 + t * 16 + l15] = acc[t][i];
    __syncthreads();
    float cm = -1e30f;
    for (int j = s; j < 512; j += 16) cm = fmaxf(cm, sc[row][j]);
    red[row][s] = cm;
    __syncthreads();
    if (s == 0) {
      float mm = rm[row];
      for (int t = 0; t < 16; ++t) mm = fmaxf(mm, red[row][t]);
      mnew[row] = mm;
    }
    __syncthreads();
    float mm = mnew[row];
    float ps = 0.f;
    for (int j = s; j < 512; j += 16) ps += __expf(sc[row][j] - mm);
    red[row][s] = ps;
    __syncthreads();
    if (s == 0) {
      float ss = 0.f;
      for (int t = 0; t < 16; ++t) ss += red[row][t];
      rl[row] = rl[row] * __expf(rm[row] - mm) + ss;
      rm[row] = mm;
    }
    __syncthreads();
  }
  if (tid < 16) {
    m3[h * 512 + r0 + tid] = rm[tid];
    l3[h * 512 + r0 + tid] = rl[tid];
  }
}

__global__ __launch_bounds__(256)
void a3v_kernel(const float* __restrict__ ql, const float* __restrict__ qkv,
                const float* __restrict__ m3, const float* __restrict__ l3,
                float* __restrict__ a3v) {
  __shared__ float sc[16][513];
  int h = blockIdx.y, rb = blockIdx.x;
  int tid = threadIdx.x, lane = tid & 31, wv = tid >> 5;
  int l15 = lane & 15, hi = lane >> 4;
  const float* qh = ql + (size_t)h * 32768;
  int r0 = rb * 16;
  int row = tid >> 4, s = tid & 15;
  v8f pv[4] = {};
  for (int jc = 0; jc < NTOK; jc += 512) {
    v8f acc[4] = {};
#pragma unroll
    for (int kk = 0; kk < 64; kk += 4) {
      int kb = kk + hi * 2;
      v2f a; a.x = qh[(r0 + l15) * 64 + kb]; a.y = qh[(r0 + l15) * 64 + kb + 1];
#pragma unroll
      for (int t = 0; t < 4; ++t) {
        int col = jc + wv * 64 + t * 16 + l15;
        v2f b;
        b.x = qkv[(size_t)col * QKVLD + 512 + h * 64 + kb];
        b.y = qkv[(size_t)col * QKVLD + 512 + h * 64 + kb + 1];
        acc[t] = wmma4(a, b, acc[t]);
      }
    }
#pragma unroll
    for (int t = 0; t < 4; ++t)
#pragma unroll
      for (int i = 0; i < 8; ++i)
        sc[i + hi * 8][wv * 64 + t * 16 + l15] = acc[t][i];
    __syncthreads();
    float mm = m3[h * 512 + r0 + row];
    for (int j = s; j < 512; j += 16) sc[row][j] = __expf(sc[row][j] - mm);
    __syncthreads();
    // P(16x512 chunk) @ V(512x64); this wave takes k-slice [wv*64, wv*64+64)
#pragma unroll
    for (int t = 0; t < 4; ++t) {
#pragma unroll
      for (int kk = 0; kk < 64; kk += 4) {
        int kb = wv * 64 + kk + hi * 2;
        v2f a; a.x = sc[l15][kb]; a.y = sc[l15][kb + 1];
        int jg = jc + kb;
        int col = t * 16 + l15;
        v2f b;
        b.x = qkv[(size_t)jg * QKVLD + 1024 + h * 64 + col];
        b.y = qkv[(size_t)(jg + 1) * QKVLD + 1024 + h * 64 + col];
        pv[t] = wmma4(a, b, pv[t]);
      }
    }
    __syncthreads();
  }
  // cross-wave reduce of 8 partial 16x64 tiles via sc (viewed as [16][512])
#pragma unroll
  for (int t = 0; t < 4; ++t)
#pragma unroll
    for (int i = 0; i < 8; ++i)
      sc[i + hi * 8][wv * 64 + t * 16 + l15] = pv[t][i];
  __syncthreads();
  for (int e = tid; e < 1024; e += 256) {
    int rr = e >> 6, d = e & 63;
    float sum = 0.f;
#pragma unroll
    for (int w = 0; w < 8; ++w) sum += sc[rr][w * 64 + d];
    a3v[(size_t)h * 32768 + (size_t)(r0 + rr) * 64 + d] = sum / l3[h * 512 + r0 + rr];
  }
}

// ---------------------------------------------------------------- out = softmax(q @ kl^T) @ whd
__global__ __launch_bounds__(256)
void out_kernel(const float* __restrict__ qkv, const float* __restrict__ kl,
                const float* __restrict__ whd, float* __restrict__ oat) {
  __shared__ float sc[16][513];
  __shared__ float red[16][16];
  __shared__ float rm[16], rl[16];
  int h = blockIdx.y, tb = blockIdx.x;
  int tid = threadIdx.x, lane = tid & 31, wv = tid >> 5;
  int l15 = lane & 15, hi = lane >> 4;
  const float* kh = kl + (size_t)h * 32768;
  int r0 = tb * 16;
  int row = tid >> 4, s = tid & 15;
  v8f acc[4] = {};
#pragma unroll
  for (int kk = 0; kk < 64; kk += 4) {
    int kb = kk + hi * 2;
    v2f a;
    a.x = qkv[(size_t)(r0 + l15) * QKVLD + h * 64 + kb];
    a.y = qkv[(size_t)(r0 + l15) * QKVLD + h * 64 + kb + 1];
#pragma unroll
    for (int t = 0; t < 4; ++t) {
      int col = wv * 64 + t * 16 + l15;
      v2f b; b.x = kh[col * 64 + kb]; b.y = kh[col * 64 + kb + 1];
      acc[t] = wmma4(a, b, acc[t]);
    }
  }
#pragma unroll
  for (int t = 0; t < 4; ++t)
#pragma unroll
    for (int i = 0; i < 8; ++i)
      sc[i + hi * 8][wv * 64 + t * 16 + l15] = acc[t][i];
  __syncthreads();
  float m = -1e30f;
  for (int j = s; j < 512; j += 16) m = fmaxf(m, sc[row][j]);
  red[row][s] = m;
  __syncthreads();
  if (s == 0) { float mm = -1e30f; for (int t = 0; t < 16; ++t) mm = fmaxf(mm, red[row][t]); rm[row] = mm; }
  __syncthreads();
  float mm = rm[row];
  float ps = 0.f;
  for (int j = s; j < 512; j += 16) { float e = __expf(sc[row][j] - mm); sc[row][j] = e; ps += e; }
  red[row][s] = ps;
  __syncthreads();
  if (s == 0) { float ss = 0.f; for (int t = 0; t < 16; ++t) ss += red[row][t]; rl[row] = 1.0f / ss; }
  __syncthreads();
  // P(16x512) @ whd(512x64)
  v8f pv[4] = {};
#pragma unroll
  for (int t = 0; t < 4; ++t) {
#pragma unroll
    for (int kk = 0; kk < 64; kk += 4) {
      int kb = wv * 64 + kk + hi * 2;
      v2f a; a.x = sc[l15][kb]; a.y = sc[l15][kb + 1];
      int col = t * 16 + l15;
      v2f b;
      b.x = whd[(size_t)h * 32768 + (size_t)kb * 64 + col];
      b.y = whd[(size_t)h * 32768 + (size_t)(kb + 1) * 64 + col];
      pv[t] = wmma4(a, b, pv[t]);
    }
  }
  __syncthreads();
#pragma unroll
  for (int t = 0; t < 4; ++t)
#pragma unroll
    for (int i = 0; i < 8; ++i)
      sc[i + hi * 8][wv * 64 + t * 16 + l15] = pv[t][i];
  __syncthreads();
  for (int e = tid; e < 1024; e += 256) {
    int rr = e >> 6, d = e & 63;
    float sum = 0.f;
#pragma unroll
    for (int w = 0; w < 8; ++w) sum += sc[rr][w * 64 + d];
    oat[(size_t)(r0 + rr) * 512 + h * 64 + d] = sum * rl[rr];
  }
}

// ---------------------------------------------------------------- depthwise conv residual
__global__ void conv_res_kernel(const float* __restrict__ qkv, const float* __restrict__ wres,
                                float* __restrict__ oat) {
  int gid = blockIdx.x * 256 + threadIdx.x;
  if (gid >= HEADS * NTOK * DHEAD) return;
  int d = gid & 63;
  int rest = gid >> 6;
  int i = rest & (NTOK - 1);
  int h = rest >> 14;
  float s = 0.f;
#pragma unroll
  for (int t = 0; t < KW; ++t) {
    int j = i + t - KW / 2;
    if (j >= 0 && j < NTOK)
      s += wres[h * KW + t] * qkv[(size_t)j * QKVLD + 1024 + h * 64 + d];
  }
  oat[(size_t)i * 512 + h * 64 + d] += s;
}

// ---------------------------------------------------------------- launch
extern "C" void kernel_launch(void* const* d_in, const int* in_sizes, int n_in,
                              void* d_out, int out_size, void* d_ws, size_t ws_size,
                              hipStream_t stream) {
  const float* x      = (const float*)d_in[0];
  const int*   labels = (const int*)d_in[1];
  const float* wqkv   = (const float*)d_in[2];
  const float* wres   = (const float*)d_in[3];
  const float* wout   = (const float*)d_in[4];
  const float* bout   = (const float*)d_in[5];
  float* out = (float*)d_out;

  char* p = (char*)d_ws;
  auto alloc = [&](size_t bytes) -> void* {
    void* r = (void*)p;
    p += (bytes + 255) & ~(size_t)255;
    return r;
  };
  int*   idx    = (int*)alloc((size_t)NTOK * 4);
  int*   counts = (int*)alloc((size_t)NCLUST * 4);
  int*   offs   = (int*)alloc((size_t)NCLUST * 4);
  float* scal   = (float*)alloc(8);
  float* qkv    = (float*)alloc((size_t)NTOK * QKVLD * 4);
  float* ql     = (float*)alloc((size_t)HEADS * 32768 * 4);
  float* kl     = (float*)alloc((size_t)HEADS * 32768 * 4);
  float* a2     = (float*)alloc((size_t)HEADS * 262144 * 4);
  float* Z      = (float*)alloc((size_t)HEADS * 262144 * 4);
  float* Z2     = (float*)alloc((size_t)HEADS * 262144 * 4);
  float* XZ     = (float*)alloc((size_t)HEADS * 262144 * 4);
  float* T0     = (float*)alloc((size_t)HEADS * 262144 * 4);
  float* T1     = (float*)alloc((size_t)HEADS * 262144 * 4);
  float* m3     = (float*)alloc((size_t)HEADS * 512 * 4);
  float* l3     = (float*)alloc((size_t)HEADS * 512 * 4);
  float* a3v    = (float*)alloc((size_t)HEADS * 32768 * 4);
  float* whd    = (float*)alloc((size_t)HEADS * 32768 * 4);
  float* oat    = (float*)alloc((size_t)NTOK * 512 * 4);

  hipMemsetAsync(counts, 0, NCLUST * 4, stream);
  hipMemsetAsync(scal, 0, 8, stream);

  hist_kernel<<<64, 256, 0, stream>>>(labels, counts);
  scan_kernel<<<1, 512, 0, stream>>>(counts, offs);
  stable_sort_kernel<<<2, 256, 0, stream>>>(labels, offs, idx);

  // qkv = x[idx] @ wqkv^T : M=16384, N=1536, K=512
  gemm_big<<<dim3(24, 128), 256, 0, stream>>>(x, DIM, idx, wqkv, qkv, QKVLD,
                                              nullptr, nullptr, DIM);

  landmarks_kernel<<<dim3(NCLUST, HEADS), 64, 0, stream>>>(qkv, offs, counts, ql, kl);
  a2_kernel<<<dim3(32, HEADS), 256, 0, stream>>>(ql, kl, a2);

  rowsum_max<<<dim3(NCLUST, HEADS), 256, 0, stream>>>(a2, scal + 0);
  colsum_max<<<dim3(NCLUST, HEADS), 256, 0, stream>>>(a2, scal + 1);
  zinit_kernel<<<dim3(1024, HEADS), 256, 0, stream>>>(a2, scal, Z);

  const long S = 262144;
  float* Zc = Z;
  float* Zn = Z2;
  for (int it = 0; it < 6; ++it) {
    gemm512<<<dim3(8, 4, HEADS), 256, 0, stream>>>(a2, 512, S, Zc, 512, S, XZ, 512, S, 1.0f, 0.0f);
    elem7i_kernel<<<dim3(1024, HEADS), 256, 0, stream>>>(XZ, T0);
    gemm512<<<dim3(8, 4, HEADS), 256, 0, stream>>>(XZ, 512, S, T0, 512, S, T1, 512, S, -1.0f, 15.0f);
    gemm512<<<dim3(8, 4, HEADS), 256, 0, stream>>>(XZ, 512, S, T1, 512, S, T0, 512, S, -1.0f, 13.0f);
    gemm512<<<dim3(8, 4, HEADS), 256, 0, stream>>>(Zc, 512, S, T0, 512, S, Zn, 512, S, 0.25f, 0.0f);
    float* t = Zc; Zc = Zn; Zn = t;
  }

  sim3_stats_kernel<<<dim3(32, HEADS), 256, 0, stream>>>(ql, qkv, m3, l3);
  a3v_kernel<<<dim3(32, HEADS), 256, 0, stream>>>(ql, qkv, m3, l3, a3v);

  // whd = a2_inv @ a3v : M=512, N=64, K=512 per head
  gemm512<<<dim3(1, 4, HEADS), 256, 0, stream>>>(Zc, 512, S, a3v, 64, 32768, whd, 64, 32768, 1.0f, 0.0f);

  out_kernel<<<dim3(1024, HEADS), 256, 0, stream>>>(qkv, kl, whd, oat);
  conv_res_kernel<<<32768, 256, 0, stream>>>(qkv, wres, oat);

  // final: out[idx[r]] = oat[r] @ wout^T + bout : M=16384, N=512, K=512
  gemm_big<<<dim3(8, 128), 256, 0, stream>>>(oat, 512, nullptr, wout, out, DIM,
                                             idx, bout, DIM);
}